// Block_45810121179249
// MI455X (gfx1250) — compile-verified
//
#include <hip/hip_runtime.h>

// ---------------------------------------------------------------------------
// MI455X (gfx1250) fp32 transformer block: GEMM + flash-attention + LayerNorm
// using V_WMMA_F32_16X16X4_F32 (wave32 layouts per CDNA5 ISA 7.12.2).
// Round 2: softmax row-sum folded into the matrix pipe via a P @ ones WMMA
// accumulator; global_prefetch for the K/V streams.
// ---------------------------------------------------------------------------

typedef __attribute__((ext_vector_type(2))) float v2f;
typedef __attribute__((ext_vector_type(8))) float v8f;

__device__ __forceinline__ v8f wmma_f32(v2f a, v2f b, v8f c) {
  // 8 args: (neg_a, A, neg_b, B, c_mod, C, reuse_a, reuse_b)
  return __builtin_amdgcn_wmma_f32_16x16x4_f32(false, a, false, b, (short)0, c,
                                               false, false);
}

constexpr int E = 256;
constexpr int S = 4096;
constexpr int BATCH = 4;
constexpr int M = BATCH * S;  // 16384 rows

// ---------------------------------------------------------------------------
// Y[m,n] = sum_k X[m,k] * W[n,k] + bias[n]   (torch Linear: x @ W.T + b)
// One wave computes one 16x16 tile; K=256 swept in 64 wmma steps.
// A-frag: lane(0..15)=row M, hi half selects K pair (K = 2*hi + v).
// B-frag: lane(0..15)=col N, K = 2*hi + v  -> contiguous float2 of W rows.
// ---------------------------------------------------------------------------
__global__ __launch_bounds__(256) void gemm16_kernel(
    const float* __restrict__ X, const float* __restrict__ W,
    const float* __restrict__ bias, float* __restrict__ Y) {
  const int lane = threadIdx.x & 31;
  const int gw = (blockIdx.x * blockDim.x + threadIdx.x) >> 5;
  const int tn = gw & 15;   // E/16 = 16 tiles along N
  const int tm = gw >> 4;   // M/16 tiles along M
  const int row = lane & 15;
  const int hi = lane >> 4;

  const float* Xr = X + (size_t)(tm * 16 + row) * E;
  const float* Wr = W + (size_t)(tn * 16 + row) * E;

  v8f acc = {};
#pragma unroll 8
  for (int k0 = 0; k0 < E; k0 += 4) {
    v2f a = *(const v2f*)(Xr + k0 + 2 * hi);
    v2f b = *(const v2f*)(Wr + k0 + 2 * hi);
    acc = wmma_f32(a, b, acc);
  }

  // C layout: element v <-> row (v + 8*hi), col = lane&15
  const int col = tn * 16 + row;
  const float bn = bias[col];
  const int m0 = tm * 16 + 8 * hi;
#pragma unroll
  for (int v = 0; v < 8; ++v)
    Y[(size_t)(m0 + v) * E + col] = acc[v] + bn;
}

// ---------------------------------------------------------------------------
// Flash attention, fp32. One wave owns (16 q-rows) x (128 output cols).
// Two waves per q-tile (eh = 0/1) each recompute the cheap QK^T scores to
// keep the P*V accumulator at 8 x v8f = 64 VGPRs/lane.
// scale = 1/sqrt(S) = 1/64 (faithful to reference's sqrt(seq_len)).
// Softmax denominator is accumulated as an extra WMMA column (P @ ones),
// which receives the same online-rescale as the output accumulators.
// ---------------------------------------------------------------------------
__global__ __launch_bounds__(128) void flash_attn_kernel(
    const float* __restrict__ Q, const float* __restrict__ K,
    const float* __restrict__ V, float* __restrict__ O) {
  __shared__ float pbuf[4][16 * 16];  // per-wave P transpose buffer

  const int lane = threadIdx.x & 31;
  const int wib = threadIdx.x >> 5;
  float* pl = pbuf[wib];

  const int gw = (blockIdx.x * 128 + threadIdx.x) >> 5;  // 2048 waves total
  const int eh = gw & 1;                 // which 128-col half of E
  const int qt = (gw >> 1) & (S / 16 - 1);
  const int b = gw >> 9;

  const int row = lane & 15;
  const int hi = lane >> 4;

  const float* Qr = Q + ((size_t)b * S + qt * 16 + row) * E;
  const float* Kb = K + (size_t)b * S * E;
  const float* Vb = V + (size_t)b * S * E;

  v8f acc[8];
#pragma unroll
  for (int j = 0; j < 8; ++j) acc[j] = (v8f){};
  v8f accs = {};  // running softmax denominator (P @ ones column)

  float rmax[8];
#pragma unroll
  for (int e = 0; e < 8; ++e) rmax[e] = -3.0e38f;

  const v2f vone = {1.0f, 1.0f};

  for (int kt = 0; kt < S / 16; ++kt) {
    const float* Kr = Kb + (size_t)(kt * 16 + row) * E;
    const float* Vr0 =
        Vb + (size_t)(kt * 16 + 2 * hi) * E + eh * 128 + row;

    // prefetch next tile's K row / V rows (speculative, safe past the end)
    __builtin_prefetch(Kr + (size_t)16 * E, 0, 3);
    __builtin_prefetch(Vr0 + (size_t)16 * E, 0, 3);

    // ---- scores: S16x16 = Q(16x256) . K_tile(16x256)^T ----
    v8f s = {};
#pragma unroll 8
    for (int k0 = 0; k0 < E; k0 += 4) {
      v2f a = *(const v2f*)(Qr + k0 + 2 * hi);
      v2f bb = *(const v2f*)(Kr + k0 + 2 * hi);
      s = wmma_f32(a, bb, s);
    }

    // ---- online softmax max-tracking. Row r = e + 8*hi lives in the 16
    // lanes of one half-wave; shfl_xor masks 1,2,4,8 reduce within it. ----
    float p[8];
#pragma unroll
    for (int e = 0; e < 8; ++e) {
      float sv = s[e] * 0.015625f;  // * 1/64
      float tmx = sv;
      for (int msk = 1; msk <= 8; msk <<= 1)
        tmx = fmaxf(tmx, __shfl_xor(tmx, msk, 32));
      float nm = fmaxf(rmax[e], tmx);
      float corr = __expf(rmax[e] - nm);
      p[e] = __expf(sv - nm);
      rmax[e] = nm;
      accs[e] *= corr;
#pragma unroll
      for (int j = 0; j < 8; ++j) acc[j][e] *= corr;
    }

    // ---- transpose P (C layout -> A-frag layout) through LDS ----
#pragma unroll
    for (int e = 0; e < 8; ++e) pl[(e + 8 * hi) * 16 + row] = p[e];
    __asm__ volatile("s_wait_dscnt 0" ::: "memory");
    v2f pa[4];
#pragma unroll
    for (int kk = 0; kk < 4; ++kk)
      pa[kk] = *(const v2f*)(pl + row * 16 + kk * 4 + 2 * hi);

    // ---- softmax denominator: accs += P @ ones (same rescale as acc) ----
#pragma unroll
    for (int kk = 0; kk < 4; ++kk) accs = wmma_f32(pa[kk], vone, accs);

    // ---- accumulate P(16x16) . V_tile(16x128-half) ----
#pragma unroll
    for (int j = 0; j < 8; ++j) {
      const float* vp = Vr0 + j * 16;
#pragma unroll
      for (int kk = 0; kk < 4; ++kk) {
        v2f bb;
        bb[0] = vp[(size_t)(kk * 4 + 0) * E];
        bb[1] = vp[(size_t)(kk * 4 + 1) * E];
        acc[j] = wmma_f32(pa[kk], bb, acc[j]);
      }
    }
  }

  // ---- normalize and store ----
  float inv[8];
#pragma unroll
  for (int e = 0; e < 8; ++e) inv[e] = 1.0f / accs[e];
  float* Or = O + ((size_t)b * S + qt * 16 + 8 * hi) * E + eh * 128 + row;
#pragma unroll
  for (int j = 0; j < 8; ++j)
#pragma unroll
    for (int e = 0; e < 8; ++e)
      Or[(size_t)e * E + j * 16] = acc[j][e] * inv[e];
}

// ---------------------------------------------------------------------------
// out = LayerNorm(A + B) * g + beta, one wave per 256-wide row.
// ---------------------------------------------------------------------------
__global__ __launch_bounds__(256) void add_ln_kernel(
    const float* __restrict__ A, const float* __restrict__ Bv,
    const float* __restrict__ g, const float* __restrict__ beta,
    float* __restrict__ Out) {
  const int lane = threadIdx.x & 31;
  const int rowid = (blockIdx.x * blockDim.x + threadIdx.x) >> 5;
  const float* pa = A + (size_t)rowid * E;
  const float* pb = Bv + (size_t)rowid * E;

  float x[8], sum = 0.0f, sq = 0.0f;
#pragma unroll
  for (int i = 0; i < 8; ++i) {
    x[i] = pa[lane + i * 32] + pb[lane + i * 32];
    sum += x[i];
    sq += x[i] * x[i];
  }
#pragma unroll
  for (int msk = 1; msk < 32; msk <<= 1) {
    sum += __shfl_xor(sum, msk, 32);
    sq += __shfl_xor(sq, msk, 32);
  }
  const float mu = sum * (1.0f / E);
  const float var = sq * (1.0f / E) - mu * mu;
  const float rs = rsqrtf(var + 1e-5f);
#pragma unroll
  for (int i = 0; i < 8; ++i) {
    const int c = lane + i * 32;
    Out[(size_t)rowid * E + c] = (x[i] - mu) * rs * g[c] + beta[c];
  }
}

// ---------------------------------------------------------------------------
extern "C" void kernel_launch(void* const* d_in, const int* in_sizes, int n_in,
                              void* d_out, int out_size, void* d_ws,
                              size_t ws_size, hipStream_t stream) {
  const float* x = (const float*)d_in[0];
  const float* Wq1 = (const float*)d_in[1];
  const float* bq1 = (const float*)d_in[2];
  const float* Wk1 = (const float*)d_in[3];
  const float* bk1 = (const float*)d_in[4];
  const float* Wv1 = (const float*)d_in[5];
  const float* bv1 = (const float*)d_in[6];
  const float* Wq2 = (const float*)d_in[7];
  const float* bq2 = (const float*)d_in[8];
  const float* Wk2 = (const float*)d_in[9];
  const float* bk2 = (const float*)d_in[10];
  const float* Wv2 = (const float*)d_in[11];
  const float* bv2 = (const float*)d_in[12];
  const float* g1 = (const float*)d_in[13];
  const float* beta1 = (const float*)d_in[14];
  const float* g2 = (const float*)d_in[15];
  const float* beta2 = (const float*)d_in[16];
  float* out = (float*)d_out;

  const size_t NE = (size_t)M * E;  // 4M floats = 16 MB
  float* ws = (float*)d_ws;
  float* q = ws;
  float* k = ws + NE;
  float* v = ws + 2 * NE;
  float* att = ws + 3 * NE;
  float* o = ws + 4 * NE;  // total 80 MB scratch

  // GEMM: (M/16)*(E/16) = 16384 waves -> 2048 blocks of 256 threads
  const dim3 gG(2048), bG(256);
  // Flash: B*(S/16)*2 = 2048 waves -> 512 blocks of 128 threads
  const dim3 gF(512), bF(128);
  // LayerNorm: 16384 row-waves -> 2048 blocks of 256
  const dim3 gL(2048), bL(256);

  // ---- layer 1 ----
  gemm16_kernel<<<gG, bG, 0, stream>>>(x, Wq1, bq1, q);
  gemm16_kernel<<<gG, bG, 0, stream>>>(x, Wk1, bk1, k);
  gemm16_kernel<<<gG, bG, 0, stream>>>(x, Wv1, bv1, v);
  flash_attn_kernel<<<gF, bF, 0, stream>>>(q, k, v, att);
  add_ln_kernel<<<gL, bL, 0, stream>>>(att, x, g1, beta1, o);

  // ---- layer 2 ----
  gemm16_kernel<<<gG, bG, 0, stream>>>(o, Wq2, bq2, q);
  gemm16_kernel<<<gG, bG, 0, stream>>>(o, Wk2, bk2, k);
  gemm16_kernel<<<gG, bG, 0, stream>>>(o, Wv2, bv2, v);
  flash_attn_kernel<<<gF, bF, 0, stream>>>(q, k, v, att);
  add_ln_kernel<<<gL, bL, 0, stream>>>(att, o, g2, beta2, out);
}